// SimpleHOGModule_41987600286418
// MI455X (gfx1250) — compile-verified
//
#include <hip/hip_runtime.h>

// ---------------------------------------------------------------------------
// 3-D HOG for MI455X (gfx1250).
// V layout in workspace: [b][z][y][x]; each sliding-window pass is a product
// with the banded matrix S[i][k] = (i <= k <= i+14).
// Pass 3 per (b,z) slice:  U = S*T  then  OUT^T = S*U^T  using
// V_WMMA_F32_16X16X4_F32 (0/1 weights -> exact f32), with one shared
// precomputed band-fragment register array feeding BOTH stages, an async
// global->LDS slice fill, and a fused transpose-store to [z,y,x,b].
// ---------------------------------------------------------------------------

#define NDIM   96
#define PLANE  (96 * 96)          // 9216
#define VOL    (96 * 96 * 96)     // 884736
#define NBINS  64

typedef float v2f __attribute__((ext_vector_type(2)));
typedef float v8f __attribute__((ext_vector_type(8)));

// ---------------------------------------------------------------------------
// Pass 1: per-voxel gradient + soft binning -> dense V[b][z][y][x]
// ---------------------------------------------------------------------------
__global__ __launch_bounds__(256) void hog_bin_kernel(const float* __restrict__ x,
                                                      float* __restrict__ V)
{
    int idx = blockIdx.x * 256 + threadIdx.x;
    if (idx >= VOL) return;
    int xg = idx % 96;
    int yg = (idx / 96) % 96;
    int zg = idx / PLANE;

    const float EPSF   = 2.220446049250313e-16f;
    const float TWO_PI = 6.283185307179586f;
    const float PI_F   = 3.141592653589793f;

    auto ld = [&](int zz, int yy, int xx) -> float {
        if ((unsigned)zz >= 96u || (unsigned)yy >= 96u || (unsigned)xx >= 96u)
            return 0.0f;                          // zero padding
        return x[(zz * 96 + yy) * 96 + xx];
    };

    float gz = ld(zg + 1, yg, xg) - ld(zg - 1, yg, xg);
    float gy = ld(zg, yg + 1, xg) - ld(zg, yg - 1, xg);
    float gx = ld(zg, yg, xg + 1) - ld(zg, yg, xg - 1);

    float r     = sqrtf(gx * gx + gy * gy + gz * gz);
    float theta = atanf(gy / (gx + EPSF));
    float phi   = acosf(gz / (r + EPSF));
    if (theta < 0.0f) theta += TWO_PI;

    float t_raw = theta / (TWO_PI / 8.0f);
    float p_raw = phi   / (PI_F  / 8.0f);

    int lt = (int)floorf(t_raw); if (lt == 8) lt = 0;
    int lp = (int)floorf(p_raw); if (lp == 8) lp = 0;
    int ht = (lt + 1) & 7;
    int hp = (lp + 1) & 7;

    float fr    = t_raw - floorf(t_raw);        // theta frac reused for phi (per reference)
    float w_lo  = fminf(fr, 1.0f - fr);
    float w_hi  = 1.0f - w_lo;
    float wp_lo = (fr == 0.0f) ? 1.0f : w_lo;
    float wp_hi = 1.0f - wp_lo;

    int   i00 = lt * 8 + lp;  float w00 = r * w_lo * wp_lo;
    int   i10 = ht * 8 + lp;  float w10 = r * w_hi * wp_lo;
    int   i01 = lt * 8 + hp;  float w01 = r * w_lo * wp_hi;
    int   i11 = ht * 8 + hp;  float w11 = r * w_hi * wp_hi;

#pragma unroll
    for (int b = 0; b < NBINS; ++b) {
        float v = 0.0f;
        v += (b == i00) ? w00 : 0.0f;
        v += (b == i10) ? w10 : 0.0f;
        v += (b == i01) ? w01 : 0.0f;
        v += (b == i11) ? w11 : 0.0f;
        V[(size_t)b * VOL + (size_t)idx] = v;   // coalesced across the wave
    }
}

// ---------------------------------------------------------------------------
// Pass 2: in-place sliding window sum along z (register-resident column scan)
// ---------------------------------------------------------------------------
__global__ __launch_bounds__(256) void hog_zsum_kernel(float* __restrict__ V)
{
    int c = blockIdx.x * 256 + threadIdx.x;     // column id over (b, y, x)
    if (c >= NBINS * PLANE) return;
    int xg = c % 96;
    int yg = (c / 96) % 96;
    int b  = c / PLANE;
    size_t base = (size_t)b * VOL + (size_t)yg * 96 + (size_t)xg;

    float v[96];
#pragma unroll
    for (int z = 0; z < 96; ++z) v[z] = V[base + (size_t)z * PLANE];

    float s = 0.0f;
#pragma unroll
    for (int z = 0; z < 15; ++z) s += v[z];

#pragma unroll
    for (int z = 0; z < 96; ++z) {
        V[base + (size_t)z * PLANE] = s;        // out[z] = sum v[z..z+14]
        s -= v[z];
        if (z + 15 < 96) s += v[z + 15];
    }
}

// ---------------------------------------------------------------------------
// Pass 3: per (b,z) slice, U = S*T then OUT^T = S*U^T with WMMA f32 16x16x4.
// 192 threads = 6 wave32; wave w owns band rows 16w..16w+15 in BOTH stages,
// so a single precomputed band-fragment array (24 x v2f) feeds every WMMA.
// Slice fill uses global_load_async_to_lds_b128 (ASYNCcnt path).
// ---------------------------------------------------------------------------
__global__ __launch_bounds__(192) void hog_yx_wmma_kernel(const float* __restrict__ V,
                                                          float* __restrict__ out)
{
    constexpr int LDP = 97;                     // padded LDS row stride (bank-friendly)
    __shared__ float Tsh[96 * LDP];             // input slice T (y major)
    __shared__ float Ush[96 * LDP];             // U = S * T

    int b = blockIdx.x / 96;
    int z = blockIdx.x % 96;
    const float* src = V + (size_t)b * VOL + (size_t)z * PLANE;

    // ---- async fill: 96 rows x 24 16B-chunks (chunks never cross a row)
    unsigned lds_base = (unsigned)(uintptr_t)(&Tsh[0]);   // low 32b = LDS byte offset
    for (int c = threadIdx.x; c < 96 * 24; c += 192) {
        int row = c / 24, col = (c % 24) * 4;
        unsigned goff = (unsigned)((row * 96 + col) * 4);
        unsigned loff = lds_base + (unsigned)((row * LDP + col) * 4);
        asm volatile("global_load_async_to_lds_b128 %0, %1, %2 offset:0"
                     :: "v"(loff), "v"(goff), "s"(src) : "memory");
        if (z < 95)                                       // warm next z-slice in L2
            __builtin_prefetch((const char*)src + PLANE * 4 + goff, 0, 1);
    }
    asm volatile("s_wait_asynccnt 0x0" ::: "memory");
    __syncthreads();

    int lane = threadIdx.x & 31;
    int wv   = threadIdx.x >> 5;                // 0..5 : band tile-row owned by this wave
    int ln   = lane & 15;                       // M of A frag / N of B frag
    int hi   = lane >> 4;                       // 0/1 -> K sub-offset 2*hi
    int mg   = 16 * wv + ln;                    // global band row for the A fragment

    // ---- precompute the banded-S A fragments for all 24 K-steps (shared by
    //      both stages): A[m][k] = (k >= m && k <= m+14), ISA 16x4 f32 layout.
    v2f sa[24];
#pragma unroll
    for (int kk = 0; kk < 24; ++kk) {
        int kg = kk * 4 + hi * 2;
        sa[kk][0] = (kg     >= mg && kg     <= mg + 14) ? 1.0f : 0.0f;
        sa[kk][1] = (kg + 1 >= mg && kg + 1 <= mg + 14) ? 1.0f : 0.0f;
    }

    // ---- Stage 1: U = S * T (window-sum over y). B[k][n] = T[k][16nt+ln].
    for (int nt = 0; nt < 6; ++nt) {
        v8f acc = {};
        for (int kk = 0; kk < 24; ++kk) {
            int kg = kk * 4 + hi * 2;
            v2f bb;
            bb[0] = Tsh[(kg    ) * LDP + 16 * nt + ln];
            bb[1] = Tsh[(kg + 1) * LDP + 16 * nt + ln];
            acc = __builtin_amdgcn_wmma_f32_16x16x4_f32(
                      false, sa[kk], false, bb, (short)0, acc, false, false);
        }
#pragma unroll
        for (int v = 0; v < 8; ++v)             // D: m = v + 8*hi (row y), n = ln (col)
            Ush[(16 * wv + v + 8 * hi) * LDP + 16 * nt + ln] = acc[v];
    }
    __syncthreads();                            // stage 2 reads all waves' U rows

    // ---- Stage 2: OUT^T = S * U^T (window-sum over x), same A fragments.
    //      B[k][n] = U^T[k][16nt+ln] = U[16nt+ln][k] -> contiguous ds_load_b64.
    float* dst = out + (size_t)z * (size_t)PLANE * NBINS + (size_t)b;
    for (int nt = 0; nt < 6; ++nt) {
        v8f acc = {};
        for (int kk = 0; kk < 24; ++kk) {
            int kg = kk * 4 + hi * 2;
            v2f bb;
            bb[0] = Ush[(16 * nt + ln) * LDP + kg    ];
            bb[1] = Ush[(16 * nt + ln) * LDP + kg + 1];
            acc = __builtin_amdgcn_wmma_f32_16x16x4_f32(
                      false, sa[kk], false, bb, (short)0, acc, false, false);
        }
#pragma unroll
        for (int v = 0; v < 8; ++v) {           // D: m = x-local, n = y-local
            int xg = 16 * wv + v + 8 * hi;
            int yg = 16 * nt + ln;
            dst[((size_t)yg * 96 + (size_t)xg) * NBINS] = acc[v];
        }
    }
}

// ---------------------------------------------------------------------------
extern "C" void kernel_launch(void* const* d_in, const int* in_sizes, int n_in,
                              void* d_out, int out_size, void* d_ws, size_t ws_size,
                              hipStream_t stream)
{
    (void)in_sizes; (void)n_in; (void)out_size; (void)ws_size;
    const float* x  = (const float*)d_in[0];
    float* out      = (float*)d_out;
    float* V        = (float*)d_ws;             // needs 64*96^3*4 = 226,492,416 B

    hog_bin_kernel    <<<VOL / 256,             256, 0, stream>>>(x, V);
    hog_zsum_kernel   <<<(NBINS * PLANE) / 256, 256, 0, stream>>>(V);
    hog_yx_wmma_kernel<<<NBINS * 96,            192, 0, stream>>>(V, out);
}